// LearnableWaveletTransform_81630148428273
// MI455X (gfx1250) — compile-verified
//
#include <hip/hip_runtime.h>
#include <stdint.h>

// MI455X (gfx1250) 3-level db4 DWT, store-bandwidth-bound (~112MB traffic).
// - conv-as-matmul via V_WMMA_F32_16X16X4_F32 (M=16 channels, N=16 outputs, K=40 banded)
// - input tiles staged via global_load_async_to_lds_b128 (ASYNCcnt path)
// - zero-padded banded-filter LDS table -> unconditional ds_load_b64 B-prep
// - LDS transpose (stride 132, conflict-free) so all global stores are float4 rows

typedef __attribute__((ext_vector_type(2))) float v2f;
typedef __attribute__((ext_vector_type(4))) float f4;
typedef __attribute__((ext_vector_type(8))) float v8f;

#define S_LEN 4096
#define F_CH  128
#define TILE  16
#define KROWS 40      // 10 K-tiles of 4 staged inputs per 16-output tile
#define OSTR  132     // transpose-tile stride: 132 % 64 == 4 -> conflict-free, 16B aligned

__constant__ float c_filt[16] = {
  // correlation kernel = reversed db4 dec_lo
  0.23037781330885523f,  0.7148465705525415f,   0.6308807679295904f,
 -0.02798376941698385f, -0.18703481171888114f,  0.030841381835986965f,
  0.032883011666982945f,-0.010597401784997278f,
  // reversed db4 dec_hi
 -0.010597401784997278f, 0.032883011666982945f, 0.030841381835986965f,
  0.18703481171888114f, -0.02798376941698385f, -0.6308807679295904f,
  0.7148465705525415f,  -0.23037781330885523f
};

template<int L_IN, int L_OUT, bool LO_FULL>
__global__ __launch_bounds__(256)
void dwt_level(const float* __restrict__ in, float* __restrict__ det,
               float* __restrict__ loout)
{
  __shared__ __align__(16) float xs[KROWS * F_CH];   // 20KB async-staged input rows
  __shared__ __align__(16) float fpad[2][72];        // zero-padded banded filters
  __shared__ __align__(16) float ot[2][TILE * OSTR]; // transposed lo/hi tiles

  const int tid = threadIdx.x;
  const int b   = blockIdx.y;
  const int o0  = blockIdx.x * TILE;
  const size_t rowbase = ((size_t)b * S_LEN + o0) * F_CH;
  const f4 zf4 = {0.f, 0.f, 0.f, 0.f};

  if (o0 >= L_OUT) {                     // pure zero-fill tile (fused pad)
    for (int q = tid; q < TILE * (F_CH / 4); q += 256) {
      const int sl = q >> 5;
      const int c4 = (q & 31) << 2;
      const size_t off = rowbase + (size_t)sl * F_CH + c4;
      *(f4*)(det + off) = zf4;
      if (LO_FULL) *(f4*)(loout + off) = zf4;
    }
    return;
  }

  // Padded banded-filter table: fpad[f][32+t] = filt[f][t], zeros elsewhere.
  // (single pass, one writer per entry -> no race, no extra barrier)
  if (tid < 144) {
    const int f = tid / 72;
    const int t = (tid % 72) - 32;
    fpad[f][tid % 72] = (t >= 0 && t < 8) ? c_filt[f * 8 + t] : 0.f;
  }

  // ---- stage 40 input rows (with zero padding) into LDS via async copy ----
  const int i0 = 2 * o0 - 6;             // left zero-pad of 6
  for (int r = 0; r < 5; ++r) {
    const int idx = (r * 256 + tid) * 4; // float index into xs (16B chunks)
    const int row = idx >> 7;
    const int col = idx & 127;
    const int ia  = i0 + row;
    if (ia >= 0 && ia < L_IN) {
      uint32_t lds  = (uint32_t)(uintptr_t)(&xs[idx]);     // LDS byte offset
      uint32_t goff = (uint32_t)((((uint32_t)b * S_LEN + (uint32_t)ia) * F_CH
                                  + (uint32_t)col) * 4u);
      asm volatile("global_load_async_to_lds_b128 %0, %1, %2"
                   :: "v"(lds), "v"(goff), "s"(in) : "memory");
    } else {
      *(f4*)(&xs[idx]) = zf4;            // zero-pad region (no async overlap)
    }
  }
#if __has_builtin(__builtin_amdgcn_s_wait_asynccnt)
  __builtin_amdgcn_s_wait_asynccnt(0);
#else
  asm volatile("s_wait_asynccnt 0" ::: "memory");
#endif
  __syncthreads();

  // ---- conv as matmul: D[ch, pos] = sum_k A[ch, k] * Band[k, pos] ----
#if __has_builtin(__builtin_amdgcn_wmma_f32_16x16x4_f32)
  {
    const int lane   = tid & 31;
    const int w      = tid >> 5;           // 8 waves = 8 groups of 16 channels
    const int lane16 = lane & 15;
    const int half   = lane >> 4;
    const int cch    = (w << 4) + lane16;  // A-matrix M (channel) per lane
    const int n2     = lane16 * 2;         // B-matrix: 2*n

    v8f alo = {0.f,0.f,0.f,0.f,0.f,0.f,0.f,0.f};
    v8f ahi = {0.f,0.f,0.f,0.f,0.f,0.f,0.f,0.f};
    #pragma unroll
    for (int kt = 0; kt < 10; ++kt) {
      const int k0 = kt * 4 + (half << 1);           // A layout: K0/K1 | K2/K3
      v2f a;
      a.x = xs[(k0    ) * F_CH + cch];
      a.y = xs[(k0 + 1) * F_CH + cch];
      // d0 = k0 - 2n is always even -> 8B-aligned pair load, no predication
      const int d0 = k0 - n2;
      const v2f blo = *(const v2f*)&fpad[0][32 + d0];
      const v2f bhi = *(const v2f*)&fpad[1][32 + d0];
      alo = __builtin_amdgcn_wmma_f32_16x16x4_f32(false, a, false, blo,
                                                  (short)0, alo, false, false);
      ahi = __builtin_amdgcn_wmma_f32_16x16x4_f32(false, a, false, bhi,
                                                  (short)0, ahi, false, false);
    }
    // D layout: lane holds N=lane16; VGPR r holds M = r + 8*half (group w).
    // r=0..7 are contiguous channels -> two b128 stores per filter.
    const int obase = lane16 * OSTR + (w << 4) + (half << 3);
    *(f4*)&ot[0][obase]     = (f4){alo[0], alo[1], alo[2], alo[3]};
    *(f4*)&ot[0][obase + 4] = (f4){alo[4], alo[5], alo[6], alo[7]};
    *(f4*)&ot[1][obase]     = (f4){ahi[0], ahi[1], ahi[2], ahi[3]};
    *(f4*)&ot[1][obase + 4] = (f4){ahi[4], ahi[5], ahi[6], ahi[7]};
  }
#else
  // VALU fallback (same math, 8-tap FMA)
  for (int q = tid; q < TILE * F_CH; q += 256) {
    const int sl  = q >> 7;
    const int cch = q & 127;
    float lo = 0.f, hi = 0.f;
    const int ib = 2 * sl;
    #pragma unroll
    for (int t = 0; t < 8; ++t) {
      const float xv = xs[(ib + t) * F_CH + cch];
      lo = fmaf(xv, fpad[0][32 + t], lo);
      hi = fmaf(xv, fpad[1][32 + t], hi);
    }
    ot[0][sl * OSTR + cch] = lo;
    ot[1][sl * OSTR + cch] = hi;
  }
#endif
  __syncthreads();

  // ---- coalesced float4 row stores (512B per row across lanes) ----
  for (int q = tid; q < TILE * (F_CH / 4); q += 256) {
    const int sl = q >> 5;
    const int c4 = (q & 31) << 2;
    const int o  = o0 + sl;
    f4 dv = zf4, lv = zf4;
    if (o < L_OUT) {
      lv = *(const f4*)&ot[0][sl * OSTR + c4];
      dv = *(const f4*)&ot[1][sl * OSTR + c4];
    }
    const size_t off = rowbase + (size_t)sl * F_CH + c4;
    *(f4*)(det + off)   = dv;
    *(f4*)(loout + off) = lv;
  }
}

// out4 = d1 + d2 + d3 (zero past row 2051); out5 = approx copy (zero past 518)
__global__ __launch_bounds__(256)
void dwt_combine(const float* __restrict__ a0, const float* __restrict__ a1,
                 const float* __restrict__ a2, const float* __restrict__ a3,
                 float* __restrict__ o4, float* __restrict__ o5)
{
  const size_t idx = (size_t)blockIdx.x * 256 + threadIdx.x;  // float4 index
  const size_t fl  = idx * 4;
  const int s = (int)((fl >> 7) & (S_LEN - 1));
  f4 r4 = {0.f,0.f,0.f,0.f};
  f4 r5 = {0.f,0.f,0.f,0.f};
  if (s < 2051) {
    const f4 x1 = ((const f4*)a1)[idx];
    const f4 x2 = ((const f4*)a2)[idx];
    const f4 x3 = ((const f4*)a3)[idx];
    r4 = x1 + x2 + x3;
  }
  if (s < 518) r5 = ((const f4*)a0)[idx];
  ((f4*)o4)[idx] = r4;
  ((f4*)o5)[idx] = r5;
}

extern "C" void kernel_launch(void* const* d_in, const int* in_sizes, int n_in,
                              void* d_out, int out_size, void* d_ws, size_t ws_size,
                              hipStream_t stream) {
  (void)in_sizes; (void)n_in; (void)out_size; (void)d_ws; (void)ws_size;
  const float* x = (const float*)d_in[0];
  float* out = (float*)d_out;
  const size_t P = (size_t)8 * S_LEN * F_CH;  // one [B,S,F] slab
  float* o0 = out;          // approx
  float* o1 = out + P;      // d1
  float* o2 = out + 2 * P;  // d2
  float* o3 = out + 3 * P;  // d3
  float* o4 = out + 4 * P;  // high_freq (used as lo1 stash first)
  float* o5 = out + 5 * P;  // low_freq  (used as lo2 stash first)

  dim3 blk(256);
  dim3 grd(S_LEN / TILE, 8);                       // full zero-fill coverage
  dwt_level<4096, 2051, false><<<grd, blk, 0, stream>>>(x,  o1, o4);
  dwt_level<2051, 1029, false><<<grd, blk, 0, stream>>>(o4, o2, o5);
  dwt_level<1029,  518, true ><<<grd, blk, 0, stream>>>(o5, o3, o0);
  dwt_combine<<<dim3((unsigned)(P / 4 / 256)), blk, 0, stream>>>(o0, o1, o2, o3,
                                                                 o4, o5);
}